// GIN_27393301414237
// MI455X (gfx1250) — compile-verified
//
#include <hip/hip_runtime.h>
#include <hip/hip_bf16.h>

typedef __attribute__((ext_vector_type(2))) float v2f;
typedef __attribute__((ext_vector_type(8))) float v8f;

static constexpr int DIM  = 96;    // feature dim (in = hid = out)
static constexpr int PSTR = 224;   // floats per K-pair row in LDS: 2*96 + 32 pad.
                                   // 224 % 64 == 32 -> the two half-waves (pair p
                                   // vs p+1) hit disjoint 32-bank sets.
static constexpr float BN_EPS = 1e-5f;

// ---------------------------------------------------------------- utilities
__global__ void fill_zero(float* __restrict__ p, int n) {
    int i = blockIdx.x * blockDim.x + threadIdx.x;
    if (i < n) p[i] = 0.0f;
}

// ----------------------------------------------------------------- degree
__global__ void gin_degree(const int* __restrict__ dst, float* __restrict__ deg, int nedges) {
    int e = blockIdx.x * blockDim.x + threadIdx.x;
    if (e < nedges) atomicAdd(&deg[dst[e]], 1.0f);
}

__global__ void gin_invdeg(float* __restrict__ deg, int n) {
    int i = blockIdx.x * blockDim.x + threadIdx.x;
    if (i < n) {
        float d = deg[i];
        deg[i] = (d > 0.0f) ? (1.0f / d) : 0.0f;
    }
}

// --------------------------------------------------------- edge scatter-add
// One wave per edge; lane L handles features L, L+32, L+64 (coalesced 128B
// bursts on the gather side, contiguous f32 atomics on the scatter side).
__global__ __launch_bounds__(256) void gin_scatter(
        const float* __restrict__ h, const int* __restrict__ src,
        const int* __restrict__ dst, float* __restrict__ agg, int nedges) {
    int wid  = (blockIdx.x * blockDim.x + threadIdx.x) >> 5;
    int lane = threadIdx.x & 31;
    if (wid >= nedges) return;
    int s = src[wid];
    int d = dst[wid];
    const float* hs = h + (long)s * DIM;
    float* ad = agg + (long)d * DIM;
#pragma unroll
    for (int j = 0; j < 3; ++j) {
        int f = lane + 32 * j;
        atomicAdd(&ad[f], hs[f]);
    }
}

// --------------------------------------------------- fused combine + GEMM
// out[M,96] = (h + inv_deg*agg)[M,96] @ W[96,96] + b   via V_WMMA_F32_16X16X4_F32
// One wave per 16-row M-tile; 6 accumulators cover all 96 output columns;
// 24 K-steps of 4. W staged in LDS K-pair interleaved:
//   Wl[(k>>1)*PSTR + col*2 + (k&1)] = W[k][col]
// so each lane's B fragment {W[ka][col], W[ka+1][col]} is one aligned
// ds_load_b64 straight into an even VGPR pair (no staging movs).
__global__ __launch_bounds__(256) void gin_gemm(
        const float* __restrict__ h, const float* __restrict__ agg,
        const float* __restrict__ invdeg, const float* __restrict__ W,
        const float* __restrict__ bias, float* __restrict__ out, int nrows) {
    __shared__ float Wl[(DIM / 2) * PSTR];
    for (int i = threadIdx.x; i < DIM * DIM; i += blockDim.x) {
        int k = i / DIM, n = i - k * DIM;
        Wl[(k >> 1) * PSTR + n * 2 + (k & 1)] = W[i];
    }
    __syncthreads();

    const int lane   = threadIdx.x & 31;
    const int wib    = threadIdx.x >> 5;                 // wave in block (0..7)
    const int mtile  = blockIdx.x * 8 + wib;
    if (mtile * 16 >= nrows) return;                     // wave-uniform: EXEC stays all-1s

    const int halfHi = lane >> 4;                        // 0: lanes 0-15, 1: lanes 16-31
    const int l16    = lane & 15;
    const int arow   = mtile * 16 + l16;                 // A-matrix row owned by this lane
    const float idg  = invdeg[arow];
    const float* hRow = h   + (long)arow * DIM;
    const float* aRow = agg + (long)arow * DIM;

    // C/D layout: VGPR r -> row (r + 8*halfHi), col = l16 + 16*t. Bias is per
    // column, so every accumulator VGPR starts at bias[col].
    v8f acc[6];
#pragma unroll
    for (int t = 0; t < 6; ++t) {
        float bv = bias[t * 16 + l16];
#pragma unroll
        for (int r = 0; r < 8; ++r) acc[t][r] = bv;
    }

    for (int k = 0; k < DIM; k += 4) {
        const int ka = k + 2 * halfHi;                   // this lane's K pair (even)
        const float2 hv = *(const float2*)(hRow + ka);
        const float2 av = *(const float2*)(aRow + ka);
        v2f a;
        a.x = hv.x + av.x * idg;                         // fused (h + mean_nb)
        a.y = hv.y + av.y * idg;
        const float* wp = Wl + (ka >> 1) * PSTR;         // K-pair row in LDS
#pragma unroll
        for (int t = 0; t < 6; ++t) {
            const int col = t * 16 + l16;
            v2f b = *(const v2f*)(wp + col * 2);         // {W[ka][col], W[ka+1][col]}
            acc[t] = __builtin_amdgcn_wmma_f32_16x16x4_f32(
                false, a, false, b, (short)0, acc[t], false, false);
        }
    }

    const int baseRow = mtile * 16 + 8 * halfHi;
#pragma unroll
    for (int t = 0; t < 6; ++t) {
        const int col = t * 16 + l16;
#pragma unroll
        for (int r = 0; r < 8; ++r)
            out[(long)(baseRow + r) * DIM + col] = acc[t][r];
    }
}

// ------------------------------------------------------------ batch-norm
// Each of the 96 threads in a block owns one feature column over a 512-row
// slab (consecutive tids -> consecutive addresses: coalesced), then one
// atomic per column per slab.
__global__ void bn_stats(const float* __restrict__ x, float* __restrict__ stats, int n) {
    int c  = threadIdx.x;                                // blockDim.x == 96
    int r0 = blockIdx.x * 512;
    int r1 = r0 + 512; if (r1 > n) r1 = n;
    float s = 0.0f, ss = 0.0f;
    for (int r = r0; r < r1; ++r) {
        float v = x[(long)r * DIM + c];
        s += v; ss += v * v;
    }
    atomicAdd(&stats[c], s);
    atomicAdd(&stats[DIM + c], ss);
}

__global__ void bn_finalize(const float* __restrict__ stats,
                            const float* __restrict__ gamma,
                            const float* __restrict__ beta,
                            float* __restrict__ scsh, float inv_n) {
    int c = threadIdx.x;
    if (c < DIM) {
        float mean = stats[c] * inv_n;
        float var  = stats[DIM + c] * inv_n - mean * mean;
        float sc   = gamma[c] * rsqrtf(var + BN_EPS);
        scsh[c]        = sc;
        scsh[DIM + c]  = beta[c] - mean * sc;
    }
}

__global__ void bn_relu(float* __restrict__ x, const float* __restrict__ scsh, int total) {
    int i = blockIdx.x * blockDim.x + threadIdx.x;
    if (i < total) {
        int c = i % DIM;
        float v = x[i] * scsh[c] + scsh[DIM + c];
        x[i] = v > 0.0f ? v : 0.0f;
    }
}

// ------------------------------------------------------------------ driver
extern "C" void kernel_launch(void* const* d_in, const int* in_sizes, int n_in,
                              void* d_out, int out_size, void* d_ws, size_t ws_size,
                              hipStream_t stream) {
    const float* features = (const float*)d_in[0];
    const float* W[3]     = {(const float*)d_in[1], (const float*)d_in[5], (const float*)d_in[9]};
    const float* bvec[3]  = {(const float*)d_in[2], (const float*)d_in[6], (const float*)d_in[10]};
    const float* gamma[2] = {(const float*)d_in[3], (const float*)d_in[7]};
    const float* beta[2]  = {(const float*)d_in[4], (const float*)d_in[8]};
    const int*   src      = (const int*)d_in[11];
    const int*   dst      = (const int*)d_in[12];

    const int N = in_sizes[0] / DIM;     // 50000
    const int E = in_sizes[11];          // 800000
    const int ND = N * DIM;

    // workspace carve-up
    float* agg    = (float*)d_ws;        // N*96
    float* H1     = agg + ND;            // N*96
    float* H2     = H1 + ND;             // N*96
    float* invdeg = H2 + ND;             // N
    float* stats  = invdeg + N;          // 192
    float* scsh   = stats + 2 * DIM;     // 192

    // degree -> inverse degree (in place)
    fill_zero<<<(N + 255) / 256, 256, 0, stream>>>(invdeg, N);
    gin_degree<<<(E + 255) / 256, 256, 0, stream>>>(dst, invdeg, E);
    gin_invdeg<<<(N + 255) / 256, 256, 0, stream>>>(invdeg, N);

    const int mtiles     = N / 16;                    // 3125 (exact)
    const int gemmBlocks = (mtiles + 7) / 8;
    const int statBlocks = (N + 511) / 512;

    const float* hin = features;
    float* outs[3] = {H1, H2, (float*)d_out};

    for (int l = 0; l < 3; ++l) {
        fill_zero<<<(ND + 255) / 256, 256, 0, stream>>>(agg, ND);
        gin_scatter<<<(E + 7) / 8, 256, 0, stream>>>(hin, src, dst, agg, E);
        gin_gemm<<<gemmBlocks, 256, 0, stream>>>(hin, agg, invdeg, W[l], bvec[l], outs[l], N);
        if (l < 2) {
            fill_zero<<<1, 256, 0, stream>>>(stats, 2 * DIM);
            bn_stats<<<statBlocks, 96, 0, stream>>>(outs[l], stats, N);
            bn_finalize<<<1, 128, 0, stream>>>(stats, gamma[l], beta[l], scsh, 1.0f / (float)N);
            bn_relu<<<(ND + 255) / 256, 256, 0, stream>>>(outs[l], scsh, ND);
        }
        hin = outs[l];
    }
}